// MonotonicThermalLUT_21509196218820
// MI455X (gfx1250) — compile-verified
//
#include <hip/hip_runtime.h>
#include <hip/hip_bf16.h>
#include <math.h>

typedef __attribute__((ext_vector_type(2))) float v2f;
typedef __attribute__((ext_vector_type(8))) float v8f;

#define BINS 2048
#define SCENE 8
#define NB 8
#define HH 1024
#define WW 1280
#define HW (HH * WW)          // 1310720 pixels per sample
#define HW4 (HW / 4)          // 327680 float4 per sample
#define EPSV 1e-8f

// ---------------------------------------------------------------------------
// order-preserving float <-> uint key transform
// ---------------------------------------------------------------------------
__device__ __forceinline__ unsigned f2key(unsigned u) {
    unsigned m = ((unsigned)((int)u >> 31)) | 0x80000000u;
    return u ^ m;
}
__device__ __forceinline__ float key2f(unsigned k) {
    unsigned u = (k & 0x80000000u) ? (k ^ 0x80000000u) : ~k;
    return __uint_as_float(u);
}

// ---------------------------------------------------------------------------
// Kernel 0: scene-mixture GEMM (8x8 @ 8x2048) via V_WMMA_F32_16X16X4_F32,
// softplus, cumsum, normalize -> lut[2048].  One wave32 block.
// ---------------------------------------------------------------------------
__global__ __launch_bounds__(32) void lut_kernel(const float* __restrict__ delta,
                                                 float* __restrict__ lut) {
    __shared__ float incr[BINS];
    __shared__ float waveTot[32];

    const int lane = threadIdx.x;        // 0..31
    const int nlo  = lane & 15;
    const int h    = lane >> 4;          // lane half

    // A matrix 16x4: scene_idx rows are all 1/SCENE (rows >=8 are don't-care,
    // filling 1/8 everywhere keeps every D row identical).
    v2f a; a[0] = 0.125f; a[1] = 0.125f;

    for (int t = 0; t < BINS / 16; ++t) {
        const int n = t * 16 + nlo;
        // B matrix 4x16 layout (ISA 7.12.2): VGPR v, lane-half h -> K = v + 2h
        v2f b0, b1;
        b0[0] = delta[(0 + 2 * h) * BINS + n];      // K = 0..3  (first WMMA)
        b0[1] = delta[(1 + 2 * h) * BINS + n];
        b1[0] = delta[(4 + 2 * h) * BINS + n];      // K = 4..7  (second WMMA)
        b1[1] = delta[(5 + 2 * h) * BINS + n];

        v8f c = {};
        c = __builtin_amdgcn_wmma_f32_16x16x4_f32(false, a, false, b0,
                                                  (short)0, c, false, false);
        c = __builtin_amdgcn_wmma_f32_16x16x4_f32(false, a, false, b1,
                                                  (short)0, c, false, false);
        // D row M=0 lives in c[0] on lanes 0..15 (lanes 16..31 hold row M=8,
        // identical by construction).
        float m  = c[0];
        float sp = fmaxf(m, 0.0f) + log1pf(expf(-fabsf(m)));  // softplus
        if (lane < 16) incr[n] = sp + EPSV;
    }
    __syncthreads();

    // cumsum of 2048 values: 64 per lane serial, then wave-scan of lane totals
    const int base = lane * 64;
    float s = 0.0f;
    for (int i = 0; i < 64; ++i) { s += incr[base + i]; incr[base + i] = s; }
    waveTot[lane] = s;
    __syncthreads();
    float run = waveTot[lane];
    for (int d = 1; d < 32; d <<= 1) {
        float nv = __shfl_up(run, d, 32);
        if (lane >= d) run += nv;
    }
    const float excl  = run - waveTot[lane];
    const float total = __shfl(run, 31, 32);
    const float scale = 2.0f / (total + EPSV);
    for (int i = 0; i < 64; ++i)
        lut[base + i] = (incr[base + i] + excl) * scale - 1.0f;
}

// ---------------------------------------------------------------------------
// Radix-select level: one workgroup (1024 threads) per sample.
// Tracks the two order statistics (rank k and k+1) independently.
// ws state per sample (8 u32): [0]=prefix_a [1]=prefix_b [2]=rank_a
// [3]=rank_b [4]=maxkey
// ---------------------------------------------------------------------------
__device__ __forceinline__ unsigned block_incl_scan(unsigned v, unsigned* wt,
                                                    int tid) {
    __syncthreads();                       // protect wt reuse between calls
    for (int d = 1; d < 32; d <<= 1) {
        unsigned nv = __shfl_up(v, d, 32);
        if ((tid & 31) >= d) v += nv;
    }
    if ((tid & 31) == 31) wt[tid >> 5] = v;
    __syncthreads();
    if (tid < 32) {
        unsigned t = wt[tid];
        for (int d = 1; d < 32; d <<= 1) {
            unsigned nv = __shfl_up(t, d, 32);
            if (tid >= d) t += nv;
        }
        wt[tid] = t;
    }
    __syncthreads();
    unsigned off = (tid >= 32) ? wt[(tid >> 5) - 1] : 0u;
    return v + off;
}

__global__ __launch_bounds__(1024) void select_kernel(
    const float4* __restrict__ x, unsigned* __restrict__ state,
    float* __restrict__ loinv, int shift, unsigned binmask, unsigned prefmask,
    int first, int last, unsigned rank0, float frac) {

    __shared__ unsigned ha[BINS];   // 2048-wide; upper half unused at level 2
    __shared__ unsigned hb[BINS];
    __shared__ unsigned wt[32];
    __shared__ unsigned smax;
    __shared__ unsigned rbin_a, rnr_a, rbin_b, rnr_b;

    const int tid = threadIdx.x;
    const int b   = blockIdx.x;
    unsigned* st  = state + b * 8;

    unsigned pref_a = 0, pref_b = 0, rank_a = rank0, rank_b = rank0 + 1;
    if (!first) {
        pref_a = st[0]; pref_b = st[1];
        rank_a = st[2]; rank_b = st[3];
    }

    for (int i = tid; i < BINS; i += 1024) { ha[i] = 0u; hb[i] = 0u; }
    if (tid == 0) smax = 0u;
    __syncthreads();

    const float4* xs = x + (size_t)b * HW4;
    unsigned lmax = 0u;
    for (int i = tid; i < HW4; i += 1024) {
        float4 v = xs[i];
        unsigned u[4] = {__float_as_uint(v.x), __float_as_uint(v.y),
                         __float_as_uint(v.z), __float_as_uint(v.w)};
        #pragma unroll
        for (int j = 0; j < 4; ++j) {
            unsigned key = f2key(u[j]);
            if (first) lmax = (key > lmax) ? key : lmax;
            if (((key ^ pref_a) & prefmask) == 0u)
                atomicAdd(&ha[(key >> shift) & binmask], 1u);
            if (((key ^ pref_b) & prefmask) == 0u)
                atomicAdd(&hb[(key >> shift) & binmask], 1u);
        }
    }
    if (first) atomicMax(&smax, lmax);
    __syncthreads();
    if (first && tid == 0) st[4] = smax;

    // locate bin containing each rank (2 bins per thread over 2048 slots)
    {
        unsigned h0 = ha[2 * tid], h1 = ha[2 * tid + 1];
        unsigned pa = h0 + h1;
        unsigned sa = block_incl_scan(pa, wt, tid);
        unsigned e0 = sa - pa, e1 = e0 + h0;
        if (rank_a >= e0 && rank_a < e1)            { rbin_a = 2 * tid;     rnr_a = rank_a - e0; }
        else if (rank_a >= e1 && rank_a < sa)       { rbin_a = 2 * tid + 1; rnr_a = rank_a - e1; }
    }
    {
        unsigned h0 = hb[2 * tid], h1 = hb[2 * tid + 1];
        unsigned pb = h0 + h1;
        unsigned sb = block_incl_scan(pb, wt, tid);
        unsigned e0 = sb - pb, e1 = e0 + h0;
        if (rank_b >= e0 && rank_b < e1)            { rbin_b = 2 * tid;     rnr_b = rank_b - e0; }
        else if (rank_b >= e1 && rank_b < sb)       { rbin_b = 2 * tid + 1; rnr_b = rank_b - e1; }
    }
    __syncthreads();

    if (tid == 0) {
        if (!last) {
            st[0] = pref_a | (rbin_a << shift);  st[2] = rnr_a;
            st[1] = pref_b | (rbin_b << shift);  st[3] = rnr_b;
        } else {
            unsigned ka = pref_a | (rbin_a << shift);
            unsigned kb = pref_b | (rbin_b << shift);
            float va = key2f(ka), vb = key2f(kb);
            float lo = va + frac * (vb - va);
            float hi = key2f(st[4]);
            loinv[2 * b]     = lo;
            loinv[2 * b + 1] = 1.0f / (hi - lo + EPSV);
        }
    }
}

// ---------------------------------------------------------------------------
// Kernel 3: normalize, LUT gather (LDS-resident LUT), 3-channel tile, f4 I/O.
// ---------------------------------------------------------------------------
__global__ __launch_bounds__(256) void apply_kernel(
    const float4* __restrict__ x, const float* __restrict__ lut,
    const float* __restrict__ loinv, float4* __restrict__ out) {

    __shared__ float slut[BINS];
    for (int i = threadIdx.x; i < BINS; i += 256) slut[i] = lut[i];
    __syncthreads();

    const int b = blockIdx.y;
    const int g = blockIdx.x * 256 + threadIdx.x;   // float4 index in sample
    const float lo  = loinv[2 * b];
    const float inv = loinv[2 * b + 1];

    float4 v = x[(size_t)b * HW4 + g];
    float4 y;
    {
        float xn = fminf(fmaxf((v.x - lo) * inv, 0.0f), 1.0f);
        y.x = slut[min(BINS - 1, (int)(xn * (float)(BINS - 1)))];
        xn = fminf(fmaxf((v.y - lo) * inv, 0.0f), 1.0f);
        y.y = slut[min(BINS - 1, (int)(xn * (float)(BINS - 1)))];
        xn = fminf(fmaxf((v.z - lo) * inv, 0.0f), 1.0f);
        y.z = slut[min(BINS - 1, (int)(xn * (float)(BINS - 1)))];
        xn = fminf(fmaxf((v.w - lo) * inv, 0.0f), 1.0f);
        y.w = slut[min(BINS - 1, (int)(xn * (float)(BINS - 1)))];
    }
    out[((size_t)b * 3 + 0) * HW4 + g] = y;
    out[((size_t)b * 3 + 1) * HW4 + g] = y;
    out[((size_t)b * 3 + 2) * HW4 + g] = y;
}

// ---------------------------------------------------------------------------
extern "C" void kernel_launch(void* const* d_in, const int* in_sizes, int n_in,
                              void* d_out, int out_size, void* d_ws, size_t ws_size,
                              hipStream_t stream) {
    const float* x     = (const float*)d_in[0];   // (8,1,1024,1280) f32
    const float* delta = (const float*)d_in[1];   // (8,2048) f32
    float*       out   = (float*)d_out;           // (8,3,1024,1280) f32

    float*    ws_lut   = (float*)d_ws;                              // 2048 f
    unsigned* ws_state = (unsigned*)((char*)d_ws + BINS * 4);       // 8*8 u32
    float*    ws_loinv = (float*)((char*)d_ws + BINS * 4 + 256);    // 16 f

    // exact interpolated-quantile position: q*(N-1)
    double pos  = 0.02 * (double)(HW - 1);
    unsigned k0 = (unsigned)pos;            // 26214
    float frac  = (float)(pos - (double)k0);

    lut_kernel<<<1, 32, 0, stream>>>(delta, ws_lut);

    const float4* x4 = (const float4*)x;
    // level 0: bits [31:21] (11 bits), also computes max key
    select_kernel<<<NB, 1024, 0, stream>>>(x4, ws_state, ws_loinv,
                                           21, 0x7FFu, 0x00000000u,
                                           1, 0, k0, frac);
    // level 1: bits [20:10] (11 bits)
    select_kernel<<<NB, 1024, 0, stream>>>(x4, ws_state, ws_loinv,
                                           10, 0x7FFu, 0xFFE00000u,
                                           0, 0, k0, frac);
    // level 2: bits [9:0] (10 bits), finalize lo / 1/(hi-lo+eps)
    select_kernel<<<NB, 1024, 0, stream>>>(x4, ws_state, ws_loinv,
                                           0, 0x3FFu, 0xFFFFFC00u,
                                           0, 1, k0, frac);

    dim3 grid(HW4 / 256, NB);
    apply_kernel<<<grid, 256, 0, stream>>>(x4, ws_lut, ws_loinv, (float4*)out);
}